// GRU_25168508355057
// MI455X (gfx1250) — compile-verified
//
#include <hip/hip_runtime.h>

// ---------------------------------------------------------------------------
// GRU forward, MI455X (gfx1250).
//   Phase 1: gi = x^T @ w_ih^T + b_ih     -> V_WMMA_F32_16X16X4_F32 tiles
//   Phase 2: sequential gated recurrence  -> single wave32,
//            GLOBAL_LOAD_ASYNC_TO_LDS_B128 double buffer (ASYNCcnt)
// ---------------------------------------------------------------------------

#define SEQ_LEN   1048576
#define IN_SZ     21
#define HID       3
#define G3        9                 // 3 * HID gate rows
#define TILES     (SEQ_LEN / 16)    // 65536 row-tiles of 16 timesteps

#define TILES_PER_WAVE   8
#define WAVES_PER_BLOCK  8
#define K1_BLOCKS (TILES / (TILES_PER_WAVE * WAVES_PER_BLOCK))  // 1024

// Async memory->LDS path (gfx1250 GLOBAL_LOAD_ASYNC_TO_LDS_B128). Set to 0 to
// fall back to VGPR-staged prefetch if the assembler rejects the mnemonic.
#define USE_ASYNC_LDS 1

typedef __attribute__((ext_vector_type(2)))  float    v2f;
typedef __attribute__((ext_vector_type(8)))  float    v8f;

// ---------------------------------------------------------------------------
// Kernel 1: gi[s][n] = sum_k x[k][s] * w_ih[n][k] + b_ih[n]   (n < 9, k < 21)
// Each wave computes 8 tiles of 16(timestep) x 16(gate; 9 live), one per loop pass.
// A (16x4 f32): lane m=lane&15 is row; lanes<16 hold K=kb+0,kb+1 in the two
// VGPRs, lanes>=16 hold K=kb+2,kb+3 (ISA 7.12.2, 32-bit A 16x4 layout).
// B (4x16 f32): VGPR r holds K=kb+r+2*half for N=lane&15.
// C/D (16x16 f32): VGPR r -> (M = r + 8*half, N = lane&15); bias in C.
// K = 21 padded to 24 -> 6 WMMAs; only chunk 5 is partial (K=20 live).
// ---------------------------------------------------------------------------
__global__ __launch_bounds__(256) void gi_wmma_kernel(
    const float* __restrict__ x,      // (21, SEQ_LEN) row-major
    const float* __restrict__ w_ih,   // (9, 21) row-major
    const float* __restrict__ b_ih,   // (9,)
    float* __restrict__ gi)           // (SEQ_LEN, 9) row-major
{
  const int lane = threadIdx.x & 31;
  const int wave = blockIdx.x * WAVES_PER_BLOCK + (threadIdx.x >> 5);
  const int half = lane >> 4;
  const int n    = lane & 15;        // tile column (gate) / tile row for A

  const float bias = (n < G3) ? b_ih[n] : 0.0f;

  // B operand per K-chunk, hoisted. Chunk 5 (K=20..23): only K=20, half==0.
  v2f bmat[6];
#pragma unroll
  for (int kc = 0; kc < 5; ++kc) {
    const int k0 = 4 * kc + 2 * half;            // <= 19 for kc<5
    bmat[kc].x = (n < G3) ? w_ih[n * IN_SZ + k0]     : 0.0f;
    bmat[kc].y = (n < G3) ? w_ih[n * IN_SZ + k0 + 1] : 0.0f;
  }
  bmat[5].x = (n < G3 && half == 0) ? w_ih[n * IN_SZ + 20] : 0.0f;
  bmat[5].y = 0.0f;

  for (int i = 0; i < TILES_PER_WAVE; ++i) {
    const int tile = wave * TILES_PER_WAVE + i;
    const float* xs = x + tile * 16 + n;         // column s = tile*16 + (lane&15)

    v8f acc;
#pragma unroll
    for (int r = 0; r < 8; ++r) acc[r] = bias;

#pragma unroll
    for (int kc = 0; kc < 5; ++kc) {             // full chunks, no predication
      const int k0 = 4 * kc + 2 * half;
      v2f a;
      a.x = xs[(size_t)k0 * SEQ_LEN];
      a.y = xs[(size_t)(k0 + 1) * SEQ_LEN];
      acc = __builtin_amdgcn_wmma_f32_16x16x4_f32(
          false, a, false, bmat[kc], (short)0, acc, false, false);
    }
    {                                            // partial chunk: K=20 only
      const float v20 = xs[(size_t)20 * SEQ_LEN]; // valid addr for all lanes
      v2f a;
      a.x = half ? 0.0f : v20;                   // v_cndmask, no exec branch
      a.y = 0.0f;
      acc = __builtin_amdgcn_wmma_f32_16x16x4_f32(
          false, a, false, bmat[5], (short)0, acc, false, false);
    }

    // One base address per lane; row stride 9 floats = 36B immediate offsets.
    float* p = gi + (size_t)(tile * 16 + 8 * half) * G3 + n;
    if (n < G3) {
#pragma unroll
      for (int r = 0; r < 8; ++r) p[r * G3] = acc[r];
    }
  }
}

// ---------------------------------------------------------------------------
// Kernel 2: sequential GRU scan + output head. One wave32 block.
// gi (36 MB) lives in the 192 MB L2 after kernel 1. Per 256-step block:
// 9216 B of gi is DMA'd memory->LDS with 18 GLOBAL_LOAD_ASYNC_TO_LDS_B128
// (one per lane-slot), double buffered; s_wait_asynccnt 18 overlaps the next
// block's copy with this block's 256 serial steps.
// ---------------------------------------------------------------------------
#define STEPS          256
#define NBLK           (SEQ_LEN / STEPS)       // 4096
#define FLOATS_PER_BLK (STEPS * G3)            // 2304
#define BYTES_PER_BLK  (FLOATS_PER_BLK * 4)    // 9216
#define V4_PER_BLK     (FLOATS_PER_BLK / 4)    // 576
#define V4_PER_LANE    (V4_PER_BLK / 32)       // 18

__device__ __forceinline__ float fexp(float x) {
  return __builtin_amdgcn_exp2f(x * 1.44269504088896340736f);   // v_exp_f32
}
__device__ __forceinline__ float frcp(float x) {
  return __builtin_amdgcn_rcpf(x);                              // v_rcp_f32
}
__device__ __forceinline__ float fsig(float x) {
  return frcp(1.0f + fexp(-x));
}
__device__ __forceinline__ float ftanh_(float x) {
  const float e = fexp(-2.0f * x);
  return (1.0f - e) * frcp(1.0f + e);
}

#if USE_ASYNC_LDS
// GVS form: global addr = SGPR64 + VGPR32 (+imm); LDS addr = VDST VGPR (+imm).
// We fold everything into the two VGPR operands, so LDS and global advance in
// lockstep (the ISA adds IOFFSET to both sides; here IOFFSET = 0).
__device__ __forceinline__ void async_copy_b128(unsigned lds_off, unsigned goff,
                                                unsigned long long gbase) {
  asm volatile("global_load_async_to_lds_b128 %0, %1, %2"
               :: "v"(lds_off), "v"(goff), "s"(gbase)
               : "memory");
}
__device__ __forceinline__ void wait_async_le18() {
  asm volatile("s_wait_asynccnt 18" ::: "memory");
}
__device__ __forceinline__ void wait_async_le0() {
  asm volatile("s_wait_asynccnt 0" ::: "memory");
}
#endif

// Single shared object => LDS offset 0; layout must match the integer LDS
// addresses handed to the async copies.
struct ScanSmem {
  float4 lbuf[2][V4_PER_BLK];   // byte offsets 0 and 9216
  float  obuf[STEPS];           // byte offset 18432
};

__global__ __launch_bounds__(32) void gru_scan_kernel(
    const float* __restrict__ gi,     // (SEQ_LEN, 9)
    const float* __restrict__ hid,    // (3,)
    const float* __restrict__ w_hh,   // (9, 3)
    const float* __restrict__ b_hh,   // (9,)
    const float* __restrict__ w_lin,  // (1, 3)
    const float* __restrict__ b_lin,  // (1,)
    float* __restrict__ out)          // (SEQ_LEN,)
{
  __shared__ ScanSmem sm;
  const int lane = threadIdx.x;            // single wave: threadIdx == lane

  // Recurrence weights -> registers (uniform, tiny).
  float whh[G3][HID], bhh[G3];
#pragma unroll
  for (int j = 0; j < G3; ++j) {
    bhh[j] = b_hh[j];
#pragma unroll
    for (int c = 0; c < HID; ++c) whh[j][c] = w_hh[j * HID + c];
  }
  const float wl0 = w_lin[0], wl1 = w_lin[1], wl2 = w_lin[2], bl = b_lin[0];
  float h0 = hid[0], h1 = hid[1], h2 = hid[2];

#if USE_ASYNC_LDS
  const unsigned long long gi_addr = (unsigned long long)gi;
  // Prime: DMA block 0 into buffer 0 (LDS base 0).
#pragma unroll
  for (int i = 0; i < V4_PER_LANE; ++i) {
    const unsigned off = (unsigned)(lane * 16 + i * 512);
    async_copy_b128(/*lds*/ off, /*goff*/ off, gi_addr);
  }
#else
  const float4* gi4 = (const float4*)gi;
  float4 stage[V4_PER_LANE];
#pragma unroll
  for (int i = 0; i < V4_PER_LANE; ++i) stage[i] = gi4[i * 32 + lane];
#endif

  for (int blk = 0; blk < NBLK; ++blk) {
#if USE_ASYNC_LDS
    if (blk + 1 < NBLK) {
      // Issue next block's DMA into the other buffer, then wait for ours:
      // async loads complete in order, so ASYNCcnt<=18 => our 18 are done.
      const unsigned long long gbase =
          gi_addr + (unsigned long long)(blk + 1) * BYTES_PER_BLK;
      const unsigned bufbase = ((blk + 1) & 1) ? (unsigned)BYTES_PER_BLK : 0u;
#pragma unroll
      for (int i = 0; i < V4_PER_LANE; ++i) {
        const unsigned off = (unsigned)(lane * 16 + i * 512);
        async_copy_b128(bufbase + off, off, gbase);
      }
      wait_async_le18();
    } else {
      wait_async_le0();
    }
    const float* g = (const float*)&sm.lbuf[blk & 1][0];
#else
    float4* dst = sm.lbuf[blk & 1];
#pragma unroll
    for (int i = 0; i < V4_PER_LANE; ++i) dst[i * 32 + lane] = stage[i];
    if (blk + 1 < NBLK) {
      const float4* src = gi4 + (size_t)(blk + 1) * V4_PER_BLK;
#pragma unroll
      for (int i = 0; i < V4_PER_LANE; ++i) stage[i] = src[i * 32 + lane];
    }
    const float* g = (const float*)dst;
#endif

    for (int t = 0; t < STEPS; ++t) {
      const float* gt = g + t * G3;

      float gh[G3];
#pragma unroll
      for (int j = 0; j < G3; ++j)
        gh[j] = bhh[j] + whh[j][0] * h0 + whh[j][1] * h1 + whh[j][2] * h2;

      const float r0 = fsig(gt[0] + gh[0]);
      const float r1 = fsig(gt[1] + gh[1]);
      const float r2 = fsig(gt[2] + gh[2]);
      const float z0 = fsig(gt[3] + gh[3]);
      const float z1 = fsig(gt[4] + gh[4]);
      const float z2 = fsig(gt[5] + gh[5]);
      const float n0 = ftanh_(gt[6] + r0 * gh[6]);
      const float n1 = ftanh_(gt[7] + r1 * gh[7]);
      const float n2 = ftanh_(gt[8] + r2 * gh[8]);
      h0 = n0 + z0 * (h0 - n0);
      h1 = n1 + z1 * (h1 - n1);
      h2 = n2 + z2 * (h2 - n2);

      const float o = fsig(h0 * wl0 + h1 * wl1 + h2 * wl2 + bl);
      if (lane == 0) sm.obuf[t] = o;     // same-wave DS ops are in-order
    }

    float* ob = out + (size_t)blk * STEPS;
#pragma unroll
    for (int i = 0; i < STEPS / 32; ++i) ob[i * 32 + lane] = sm.obuf[i * 32 + lane];
  }
}

// ---------------------------------------------------------------------------
// Host launch. Inputs (setup_inputs order):
//   0:input(21,2^20) 1:hidden(3) 2:w_ih(9,21) 3:w_hh(9,3)
//   4:b_ih(9) 5:b_hh(9) 6:w_lin(1,3) 7:b_lin(1)
// Workspace: gi = SEQ_LEN * 9 floats = 36 MB.
// ---------------------------------------------------------------------------
extern "C" void kernel_launch(void* const* d_in, const int* in_sizes, int n_in,
                              void* d_out, int out_size, void* d_ws, size_t ws_size,
                              hipStream_t stream) {
  (void)in_sizes; (void)n_in; (void)out_size; (void)ws_size;
  const float* x     = (const float*)d_in[0];
  const float* hid   = (const float*)d_in[1];
  const float* w_ih  = (const float*)d_in[2];
  const float* w_hh  = (const float*)d_in[3];
  const float* b_ih  = (const float*)d_in[4];
  const float* b_hh  = (const float*)d_in[5];
  const float* w_lin = (const float*)d_in[6];
  const float* b_lin = (const float*)d_in[7];
  float* out = (float*)d_out;
  float* gi  = (float*)d_ws;

  gi_wmma_kernel<<<K1_BLOCKS, 256, 0, stream>>>(x, w_ih, b_ih, gi);
  gru_scan_kernel<<<1, 32, 0, stream>>>(gi, hid, w_hh, b_hh, w_lin, b_lin, out);
}